// ReduceBoundingBoxes_27994596836014
// MI455X (gfx1250) — compile-verified
//
#include <hip/hip_runtime.h>
#include <stdint.h>

// ---------------------------------------------------------------------------
// ReduceBoundingBoxes (score threshold + stable descending sort + greedy NMS)
// MI455X / gfx1250 design:
//  * tiny, latency-bound problem (184KB in / 184KB out) -> LDS-resident
//  * TDM (tensor_load_to_lds) DMAs the score plane into LDS (TENSORcnt path)
//  * enumeration ranking (stable: score desc, index asc) instead of a sort
//  * single-block sequential greedy NMS with register-cached boxes and a
//    double-buffered LDS broadcast slot (1 barrier per pivot)
//  * WMMA f32 16x16x4 for the exact integer valid-count reduction (verified
//    lowering to v_wmma_f32_16x16x4_f32 in round 1)
// ---------------------------------------------------------------------------

#define NPATCH     96
#define NCELL      (NPATCH * NPATCH)        // 9216
#define WIDTH_F    3072.0f
#define HEIGHT_F   2304.0f
#define XPS_F      32.0f                    // 3072 / 96
#define YPS_F      24.0f                    // 2304 / 96
#define P_THRESH   0.9f
#define IOU_THRESH 0.5f

#define K1_BLOCK   256
#define K1_GRID    (NCELL / K1_BLOCK)       // 36
#define K2_THREADS 1024
#define K2_SLOTS   ((NCELL + K2_THREADS - 1) / K2_THREADS)   // 9

typedef __attribute__((ext_vector_type(2))) float        v2f;
typedef __attribute__((ext_vector_type(8))) float        v8f;
typedef __attribute__((ext_vector_type(4))) unsigned int u32x4;
typedef __attribute__((ext_vector_type(4))) int          i32x4;
typedef __attribute__((ext_vector_type(8))) int          i32x8;

// Hard probes apply ONLY to the device pass — AMDGCN builtins are not
// declared during the host pass (that is what broke round 2).
#if defined(__HIP_DEVICE_COMPILE__)
#  if !__has_builtin(__builtin_amdgcn_tensor_load_to_lds)
#    error "CDNA5 device probe: __builtin_amdgcn_tensor_load_to_lds not declared"
#  endif
#  if !__has_builtin(__builtin_amdgcn_s_wait_tensorcnt)
#    error "CDNA5 device probe: __builtin_amdgcn_s_wait_tensorcnt not declared"
#  endif
#  if !__has_builtin(__builtin_amdgcn_wmma_f32_16x16x4_f32)
#    error "CDNA5 device probe: __builtin_amdgcn_wmma_f32_16x16x4_f32 not declared"
#  endif
#  define USE_CDNA5_PATHS 1
#endif

// DMA the 9216-float score plane (x[0]) from global into LDS via the Tensor
// Data Mover. One wave issues the descriptor; everyone syncs after.
__device__ __forceinline__ void load_scores_to_lds(const float* __restrict__ xg,
                                                   float* sdst, int tid, int nthr)
{
#ifdef USE_CDNA5_PATHS
  (void)nthr;
  if (tid < 32) {
    // LDS aperture: generic addr low 32 bits == LDS byte offset
    unsigned lds_off = (unsigned)(uintptr_t)sdst;
    unsigned long long ga = (unsigned long long)(uintptr_t)xg;

    // D# group 0: count=1, lds_addr, global_addr[56:0], type=2 ("image")
    u32x4 g0;
    g0[0] = 1u;
    g0[1] = lds_off;
    g0[2] = (unsigned)(ga & 0xFFFFFFFFull);
    g0[3] = (unsigned)((ga >> 32) & 0x01FFFFFFull) | (2u << 30);

    // D# group 1: data_size=2 (4B), tensor_dim0=9216, tensor_dim1=1,
    // tile_dim0=9216 (1-D tile), tensor_dim0_stride=9216, no padding/mask.
    i32x8 g1;
    g1[0] = (int)(2u << 16);                          // mask=0, data_size=4B
    g1[1] = (int)(((unsigned)NCELL & 0xFFFFu) << 16); // tensor_dim0[15:0]
    g1[2] = (int)((((unsigned)NCELL >> 16) & 0xFFFFu) // tensor_dim0[31:16]
                  | (1u << 16));                      // tensor_dim1 = 1
    g1[3] = (int)(((unsigned)NCELL & 0xFFFFu) << 16); // tile_dim0 = 9216
    g1[4] = 0;                                        // tile_dim1/2 unused
    g1[5] = (int)NCELL;                               // tensor_dim0_stride
    g1[6] = 0;
    g1[7] = 0;

    i32x4 gz = {0, 0, 0, 0};
#if __has_include(<hip/amd_detail/amd_gfx1250_TDM.h>)
    i32x8 gz8 = {0, 0, 0, 0, 0, 0, 0, 0};
    __builtin_amdgcn_tensor_load_to_lds(g0, g1, gz, gz, gz8, 0);
#else
    __builtin_amdgcn_tensor_load_to_lds(g0, g1, gz, gz, 0);
#endif
    __builtin_amdgcn_s_wait_tensorcnt((short)0);
  }
#else
  // Host-pass placeholder (never executed; keeps host parse clean).
  for (int c = tid; c < NCELL; c += nthr) sdst[c] = xg[c];
#endif
  __syncthreads();
}

// Wave-level sum via WMMA on device; plain code for the host parse.
__device__ __forceinline__ float wave_sum32(float part)
{
#ifdef USE_CDNA5_PATHS
  // A[m][k]: lane L<16 -> {part,0} in K=0,1 ; lane L>=16 -> {part,0} in K=2,3.
  // B = ones(4x16) => D[m][n] = part_m + part_{m+16}; summing the 8 D regs in
  // a lane covers one half of the rows; one xor-16 shuffle completes the wave.
  v2f a; a[0] = part; a[1] = 0.0f;
  v2f b; b[0] = 1.0f; b[1] = 1.0f;
  v8f cz = {};
  v8f d = __builtin_amdgcn_wmma_f32_16x16x4_f32(false, a, false, b,
                                                (short)0, cz, false, false);
  float h = d[0] + d[1] + d[2] + d[3] + d[4] + d[5] + d[6] + d[7];
  return h + __shfl_xor(h, 16, 32);
#else
  float wsum = part;
  for (int off = 16; off > 0; off >>= 1) wsum += __shfl_xor(wsum, off, 32);
  return wsum;
#endif
}

// ---------------------------------------------------------------------------
// Kernel 1: zero output rows; for each valid cell (score > 0.9) compute its
// stable rank (score desc, index asc) among all cells by enumeration and
// scatter the cell index into ws. Invalid cells can never out-rank valid ones
// (score <= 0.9 < valid score), so no validity test in the inner scan.
// ---------------------------------------------------------------------------
__global__ __launch_bounds__(K1_BLOCK)
void k1_rank_scatter(const float* __restrict__ x, float* __restrict__ out,
                     int* __restrict__ sortedCell)
{
  __shared__ float s_sc[NCELL];
  const int tid = threadIdx.x;
  load_scores_to_lds(x, s_sc, tid, K1_BLOCK);

  const int c = blockIdx.x * K1_BLOCK + tid;
  if (c < NCELL) {
    float* o = out + c * 5;
    o[0] = 0.0f; o[1] = 0.0f; o[2] = 0.0f; o[3] = 0.0f; o[4] = 0.0f;

    const float sc = s_sc[c];
    if (sc > P_THRESH) {
      int rank = 0;
#pragma unroll 8
      for (int f = 0; f < NCELL; ++f) {
        const float sf = s_sc[f];
        rank += (int)((sf > sc) || ((sf == sc) && (f < c)));
      }
      sortedCell[rank] = c;
    }
  }
}

// ---------------------------------------------------------------------------
// Kernel 2: single block. Count valid M (WMMA reduction), cache sorted boxes
// in registers (9 slots/thread), sequential greedy NMS with double-buffered
// LDS pivot broadcast, then write kept rows (others already zeroed by k1).
// ---------------------------------------------------------------------------
__global__ __launch_bounds__(K2_THREADS)
void k2_nms(const float* __restrict__ x, const int* __restrict__ sortedCell,
            float* __restrict__ out)
{
  __shared__ float s_sc[NCELL];
  __shared__ float s_b[2][8];
  __shared__ float s_ws[K2_THREADS / 32];
  __shared__ int   s_M;

  const int tid = threadIdx.x;
  load_scores_to_lds(x, s_sc, tid, K2_THREADS);

  // ---- valid count M (exact integer arithmetic in f32) ----
  float part = 0.0f;
  for (int c = tid; c < NCELL; c += K2_THREADS)
    part += (s_sc[c] > P_THRESH) ? 1.0f : 0.0f;

  float wsum = wave_sum32(part);

  const int lane = tid & 31, wave = tid >> 5;
  if (lane == 0) s_ws[wave] = wsum;
  __syncthreads();
  if (tid == 0) {
    float t = 0.0f;
    for (int w = 0; w < K2_THREADS / 32; ++w) t += s_ws[w];
    s_M = (int)(t + 0.5f);
  }
  __syncthreads();
  const int M = s_M;

  // ---- cache my sorted boxes in registers ----
  float px1[K2_SLOTS], py1[K2_SLOTS], px2[K2_SLOTS], py2[K2_SLOTS];
  float ps[K2_SLOTS], pa[K2_SLOTS];
  bool  pk[K2_SLOTS];
#pragma unroll
  for (int k = 0; k < K2_SLOTS; ++k) {
    pk[k] = false;
    const int p = tid + k * K2_THREADS;
    if (p < M) {
      const int cc = sortedCell[p];
      const int i = cc / NPATCH;
      const int j = cc % NPATCH;
      const float v1 = x[1 * NCELL + cc];
      const float v2 = x[2 * NCELL + cc];
      const float v3 = x[3 * NCELL + cc];
      const float v4 = x[4 * NCELL + cc];
      const float fi = (float)i * XPS_F;
      const float fj = (float)j * YPS_F;
      const float X1 = v1 * WIDTH_F + fi;
      const float Y1 = v2 * HEIGHT_F + fj;
      const float X2 = (v3 - v1) * WIDTH_F + fi;
      const float Y2 = (v4 - v2) * HEIGHT_F + fj;
      px1[k] = X1; py1[k] = Y1; px2[k] = X2; py2[k] = Y2;
      ps[k]  = s_sc[cc];
      pa[k]  = (X2 - X1) * (Y2 - Y1);
      pk[k]  = true;
    }
  }

  // ---- greedy NMS: pivot i broadcast via double-buffered LDS slot ----
  for (int i = 0; i < M; ++i) {
    const int buf = i & 1;
    if (tid == (i & (K2_THREADS - 1))) {
      const int sl = i >> 10;                     // K2_THREADS == 1024
      s_b[buf][0] = pk[sl] ? 1.0f : 0.0f;
      s_b[buf][1] = px1[sl]; s_b[buf][2] = py1[sl];
      s_b[buf][3] = px2[sl]; s_b[buf][4] = py2[sl];
      s_b[buf][5] = pa[sl];
    }
    __syncthreads();
    if (s_b[buf][0] != 0.0f) {
      const float ix1 = s_b[buf][1], iy1 = s_b[buf][2];
      const float ix2 = s_b[buf][3], iy2 = s_b[buf][4];
      const float ia  = s_b[buf][5];
#pragma unroll
      for (int k = 0; k < K2_SLOTS; ++k) {
        const int p = tid + k * K2_THREADS;
        if (p > i && pk[k]) {
          float iw = fminf(ix2, px2[k]) - fmaxf(ix1, px1[k]);
          iw = fmaxf(iw, 0.0f);
          float ih = fminf(iy2, py2[k]) - fmaxf(iy1, py1[k]);
          ih = fmaxf(ih, 0.0f);
          const float inter = iw * ih;
          const float iou = inter / (ia + pa[k] - inter + 1e-9f);
          if (iou > IOU_THRESH) pk[k] = false;
        }
      }
    }
  }

  // ---- write kept rows (score, x1, y1, w, h); rest already zero ----
#pragma unroll
  for (int k = 0; k < K2_SLOTS; ++k) {
    const int p = tid + k * K2_THREADS;
    if (p < M && pk[k]) {
      float* o = out + p * 5;
      o[0] = ps[k];
      o[1] = px1[k];
      o[2] = py1[k];
      o[3] = px2[k] - px1[k];
      o[4] = py2[k] - py1[k];
    }
  }
}

// ---------------------------------------------------------------------------
extern "C" void kernel_launch(void* const* d_in, const int* in_sizes, int n_in,
                              void* d_out, int out_size, void* d_ws, size_t ws_size,
                              hipStream_t stream)
{
  (void)in_sizes; (void)n_in; (void)out_size; (void)ws_size;
  const float* x = (const float*)d_in[0];
  float* out = (float*)d_out;
  int* sortedCell = (int*)d_ws;   // 9216 * 4B scratch

  k1_rank_scatter<<<K1_GRID, K1_BLOCK, 0, stream>>>(x, out, sortedCell);
  k2_nms<<<1, K2_THREADS, 0, stream>>>(x, sortedCell, out);
}